// GNN_4183298146853
// MI455X (gfx1250) — compile-verified
//
#include <hip/hip_runtime.h>
#include <hip/hip_bf16.h>

// ---------------------------------------------------------------------------
// GraphConv x2 + mean-pool + linear head for MI455X (gfx1250, wave32).
//
// Phase breakdown / roofline:
//   edge scatter: ~650MB traffic per layer -> HBM bound (~60us floor @23.3TB/s)
//   GEMMs: 4x [20000x128]x[128x128] fp32 -> V_WMMA_F32_16X16X4_F32
// ---------------------------------------------------------------------------

#define N_NODES   20000
#define N_EDGES   640000
#define DIM       128
#define NGRAPHS   64

typedef __attribute__((ext_vector_type(2))) float v2f;
typedef __attribute__((ext_vector_type(8))) float v8f;

// ---------------------------------------------------------------------------
// Zero the accumulator regions (grid-stride).
// ---------------------------------------------------------------------------
__global__ __launch_bounds__(256) void gnn_zero_f32(float* __restrict__ p, int n) {
    int i = blockIdx.x * blockDim.x + threadIdx.x;
    int stride = gridDim.x * blockDim.x;
    for (; i < n; i += stride) p[i] = 0.0f;
}

// ---------------------------------------------------------------------------
// Edge scatter: agg[dst[e], :] += edge_attr[e] * x[src[e], :]
// One wave (32 lanes) per edge; each lane handles 4 consecutive features
// (float4 coalesced load, 4x global_atomic_add_f32).
// Block = 256 threads = 8 edges.
// ---------------------------------------------------------------------------
__global__ __launch_bounds__(256) void gnn_edge_scatter(
    const float* __restrict__ x,
    const int*   __restrict__ src,
    const int*   __restrict__ dst,
    const float* __restrict__ ew,
    float*       __restrict__ agg,
    int nEdges)
{
    int wave = threadIdx.x >> 5;
    int lane = threadIdx.x & 31;
    int e = blockIdx.x * 8 + wave;
    if (e >= nEdges) return;
    int s = src[e];
    int d = dst[e];
    float w = ew[e];
    const float4* xs = (const float4*)(x + (size_t)s * DIM);
    float4 v = xs[lane];
    float* ad = agg + (size_t)d * DIM + lane * 4;
    atomicAdd(ad + 0, v.x * w);
    atomicAdd(ad + 1, v.y * w);
    atomicAdd(ad + 2, v.z * w);
    atomicAdd(ad + 3, v.w * w);
}

// ---------------------------------------------------------------------------
// Fused GraphConv GEMM:
//   out = [relu]( agg @ Wrel^T + xin @ Wroot^T + bias )
// One block = 256 threads = 8 waves; block covers 16 rows x 128 cols.
// Each wave computes one 16x16 tile with V_WMMA_F32_16X16X4_F32, K=128.
//
// fp32 WMMA fragment layouts (ISA 7.12.2, wave32):
//   A (16x4, MxK):  lanes 0-15: M=lane, {v0,v1}={K,K+1}; lanes 16-31: {K+2,K+3}
//   B (4x16, KxN):  lanes 0-15: N=lane, {v0,v1}=rows {K,K+1}; lanes 16-31: {K+2,K+3}
//     with B[k][n] = W[colBase+n][k]  (i.e. W^T)
//   C/D (16x16):    vgpr i: lanes 0-15 -> (M=i, N=lane); lanes 16-31 -> (M=i+8)
// ---------------------------------------------------------------------------
__global__ __launch_bounds__(256) void gnn_graphconv_wmma(
    const float* __restrict__ agg,    // [N,128] aggregated messages
    const float* __restrict__ xin,    // [N,128] root features
    const float* __restrict__ Wrel,   // [128,128] row-major [out,in]
    const float* __restrict__ Wroot,  // [128,128]
    const float* __restrict__ bias,   // [128]
    float*       __restrict__ out,    // [N,128]
    int doRelu)
{
    const int wave    = threadIdx.x >> 5;    // 0..7 -> output column tile
    const int lane    = threadIdx.x & 31;
    const int half    = lane >> 4;           // 0: K,K+1   1: K+2,K+3
    const int l       = lane & 15;
    const int rowBase = blockIdx.x * 16;     // N_NODES % 16 == 0 -> no tail
    const int colBase = wave * 16;

    const float* Aagg  = agg   + (size_t)(rowBase + l) * DIM;
    const float* Ax    = xin   + (size_t)(rowBase + l) * DIM;
    const float* Brel  = Wrel  + (size_t)(colBase + l) * DIM;
    const float* Broot = Wroot + (size_t)(colBase + l) * DIM;

    v8f c = {};   // accumulate both products + add bias at store time

#pragma unroll
    for (int k = 0; k < DIM; k += 4) {
        const int ka = k + 2 * half;
        v2f a, b;
        // rel path: A = agg tile, B = Wrel^T tile
        a.x = Aagg[ka];  a.y = Aagg[ka + 1];
        b.x = Brel[ka];  b.y = Brel[ka + 1];
        c = __builtin_amdgcn_wmma_f32_16x16x4_f32(false, a, false, b,
                                                  (short)0, c, false, false);
        // root path: A = x tile, B = Wroot^T tile
        a.x = Ax[ka];    a.y = Ax[ka + 1];
        b.x = Broot[ka]; b.y = Broot[ka + 1];
        c = __builtin_amdgcn_wmma_f32_16x16x4_f32(false, a, false, b,
                                                  (short)0, c, false, false);
    }

    const float bv  = bias[colBase + l];
    const int   col = colBase + l;
#pragma unroll
    for (int i = 0; i < 8; ++i) {
        int row = rowBase + (half ? (i + 8) : i);
        float v = c[i] + bv;
        if (doRelu) v = fmaxf(v, 0.0f);
        out[(size_t)row * DIM + col] = v;
    }
}

// ---------------------------------------------------------------------------
// Mean-pool accumulation: sums[batch[n], :] += h[n, :]; counts[batch[n]] += 1
// One wave per node, float4 per lane.
// ---------------------------------------------------------------------------
__global__ __launch_bounds__(256) void gnn_pool(
    const float* __restrict__ h,
    const int*   __restrict__ batch,
    float*       __restrict__ sums,
    float*       __restrict__ counts,
    int nNodes)
{
    int wave = threadIdx.x >> 5;
    int lane = threadIdx.x & 31;
    int n = blockIdx.x * 8 + wave;
    if (n >= nNodes) return;
    int g = batch[n];
    const float4* hv = (const float4*)(h + (size_t)n * DIM);
    float4 v = hv[lane];
    float* sd = sums + (size_t)g * DIM + lane * 4;
    atomicAdd(sd + 0, v.x);
    atomicAdd(sd + 1, v.y);
    atomicAdd(sd + 2, v.z);
    atomicAdd(sd + 3, v.w);
    if (lane == 0) atomicAdd(counts + g, 1.0f);
}

// ---------------------------------------------------------------------------
// Head: out[g] = relu( dot(sums[g]/max(cnt,1), Wlin) + blin )
// One wave per graph; wave32 shfl_xor reduction.
// ---------------------------------------------------------------------------
__global__ __launch_bounds__(256) void gnn_head(
    const float* __restrict__ sums,
    const float* __restrict__ counts,
    const float* __restrict__ Wlin,
    const float* __restrict__ blin,
    float*       __restrict__ outv)
{
    int wave = threadIdx.x >> 5;
    int lane = threadIdx.x & 31;
    int g = blockIdx.x * 8 + wave;
    if (g >= NGRAPHS) return;
    const float4* sv = (const float4*)(sums + (size_t)g * DIM);
    const float4* wv = (const float4*)Wlin;
    float4 s = sv[lane];
    float4 w = wv[lane];
    float p = s.x * w.x + s.y * w.y + s.z * w.z + s.w * w.w;
#pragma unroll
    for (int off = 16; off > 0; off >>= 1) p += __shfl_xor(p, off, 32);
    if (lane == 0) {
        float cnt = fmaxf(counts[g], 1.0f);
        float r = p / cnt + blin[0];
        outv[g] = fmaxf(r, 0.0f);
    }
}

// ---------------------------------------------------------------------------
extern "C" void kernel_launch(void* const* d_in, const int* in_sizes, int n_in,
                              void* d_out, int out_size, void* d_ws, size_t ws_size,
                              hipStream_t stream) {
    const float* x        = (const float*)d_in[0];           // [20000,128]
    const int*   edge_idx = (const int*)  d_in[1];           // [2,640000]
    const int*   batch    = (const int*)  d_in[2];           // [20000]
    const float* eattr    = (const float*)d_in[3];           // [640000]
    const float* W1_rel   = (const float*)d_in[4];
    const float* b1       = (const float*)d_in[5];
    const float* W1_root  = (const float*)d_in[6];
    const float* W3_rel   = (const float*)d_in[7];
    const float* b3       = (const float*)d_in[8];
    const float* W3_root  = (const float*)d_in[9];
    const float* Wlin     = (const float*)d_in[10];
    const float* blin     = (const float*)d_in[11];
    float* out = (float*)d_out;

    const int* src = edge_idx;             // edge_index[0]
    const int* dst = edge_idx + N_EDGES;   // edge_index[1]

    // Workspace layout: accumulators first (contiguous -> single zero pass).
    const size_t NH = (size_t)N_NODES * DIM;            // 2,560,000 floats
    float* agg1   = (float*)d_ws;                       // [N,128]
    float* agg2   = agg1 + NH;                          // [N,128]
    float* sums   = agg2 + NH;                          // [64,128]
    float* counts = sums + NGRAPHS * DIM;               // [64]
    float* h1     = counts + NGRAPHS;                   // [N,128]
    float* h2     = h1 + NH;                            // [N,128]

    const int nZero = (int)(2 * NH + NGRAPHS * DIM + NGRAPHS);

    // 1) zero accumulators (every call: deterministic replays)
    gnn_zero_f32<<<2048, 256, 0, stream>>>(agg1, nZero);

    // 2) layer-1 scatter: agg1 += edge_attr * x[src]
    gnn_edge_scatter<<<N_EDGES / 8, 256, 0, stream>>>(x, src, dst, eattr, agg1, N_EDGES);

    // 3) h1 = relu(agg1 @ W1_rel^T + x @ W1_root^T + b1)   [WMMA]
    gnn_graphconv_wmma<<<N_NODES / 16, 256, 0, stream>>>(agg1, x, W1_rel, W1_root, b1, h1, 1);

    // 4) layer-3 scatter: agg2 += edge_attr * h1[src]
    gnn_edge_scatter<<<N_EDGES / 8, 256, 0, stream>>>(h1, src, dst, eattr, agg2, N_EDGES);

    // 5) h2 = agg2 @ W3_rel^T + h1 @ W3_root^T + b3        [WMMA]
    gnn_graphconv_wmma<<<N_NODES / 16, 256, 0, stream>>>(agg2, h1, W3_rel, W3_root, b3, h2, 0);

    // 6) mean-pool accumulation
    gnn_pool<<<N_NODES / 8, 256, 0, stream>>>(h2, batch, sums, counts, N_NODES);

    // 7) head
    gnn_head<<<NGRAPHS / 8, 256, 0, stream>>>(sums, counts, Wlin, blin, out);

    (void)in_sizes; (void)n_in; (void)out_size; (void)ws_size;
}